// GeneratorQuantumCircuit_29128468201911
// MI455X (gfx1250) — compile-verified
//
#include <hip/hip_runtime.h>
#include <hip/hip_bf16.h>

typedef float v2f __attribute__((ext_vector_type(2)));
typedef float v8f __attribute__((ext_vector_type(8)));

#define N_QUBITS 4
#define N_LAYERS 2
#define TILES_PER_WAVE 8
#define WAVES_PER_BLOCK 8

// SWAPX16 lane swizzle: group-of-32, xor_mask=0x10, and_mask=0x1f
#define SWIZZLE_SWAPX16 0x401f
__device__ __forceinline__ float swap_half_f32(float x) {
    return __int_as_float(
        __builtin_amdgcn_ds_swizzle(__float_as_int(x), SWIZZLE_SWAPX16));
}

// ---------------------------------------------------------------------------
// Pre-kernel: build the fixed 16x16 real matrix V = product of the CNOT-ring +
// RY(weights) layers. Column j = circuit applied to basis vector e_j.
// Wire w corresponds to bit position (3-w) in the flat index.
// ---------------------------------------------------------------------------
__global__ void build_V_kernel(const float* __restrict__ weights,
                               float* __restrict__ V) {
    const int j = threadIdx.x;
    if (j >= 16) return;
    float st[16];
#pragma unroll
    for (int m = 0; m < 16; ++m) st[m] = (m == j) ? 1.0f : 0.0f;

#pragma unroll
    for (int layer = 0; layer < N_LAYERS; ++layer) {
#pragma unroll
        for (int w = 0; w < 4; ++w) {  // CNOT(w, (w+1)%4) in sequence
            const int cb = 1 << (3 - w);
            const int tb = 1 << (3 - ((w + 1) & 3));
            float ns[16];
#pragma unroll
            for (int m = 0; m < 16; ++m)
                ns[m] = (m & cb) ? st[m ^ tb] : st[m];
#pragma unroll
            for (int m = 0; m < 16; ++m) st[m] = ns[m];
        }
#pragma unroll
        for (int w = 0; w < 4; ++w) {  // RY(weights[layer][w])
            const float th = 0.5f * weights[layer * 4 + w];
            float s, c;
            __sincosf(th, &s, &c);
            const int b = 1 << (3 - w);
#pragma unroll
            for (int m = 0; m < 16; ++m) {
                if (!(m & b)) {
                    const float a0 = st[m], a1 = st[m | b];
                    st[m]     = c * a0 - s * a1;
                    st[m | b] = s * a0 + c * a1;
                }
            }
        }
    }
#pragma unroll
    for (int m = 0; m < 16; ++m) V[m * 16 + j] = st[m];  // row-major V[m][j]
}

// ---------------------------------------------------------------------------
// Main kernel: 16 batch elements per wave-tile.
// Per lane B-fragment slots cover j = 4c + 2*half + d; amplitude
// amp_j = (-i)^pc(j) * r_j is purely real or imaginary, with placement/sign
// fully determined by q = popcount(c)+d (compile-time) and `half` (runtime):
//   re sign over q=0..3 : +,-,-,+ (active when half == q&1)
//   im sign over q=0..3 : -,-,+,+ (active when half != q&1)
// D_re = V*S_re, D_im = V*S_im via 8x V_WMMA_F32_16X16X4_F32; probs and
// marginals reduce in-register + one SWAPX16 swizzle per output.
// ---------------------------------------------------------------------------
__global__ __launch_bounds__(256) void qcirc_wmma_kernel(
    const float* __restrict__ noise, const float* __restrict__ V,
    float* __restrict__ out, int nbatch) {
    const int lane  = threadIdx.x & 31;
    const int n16   = lane & 15;
    const int half  = lane >> 4;
    const bool hb   = (half != 0);
    const int wave  = blockIdx.x * WAVES_PER_BLOCK + (threadIdx.x >> 5);

    // Loop-invariant A fragments: lane holds V[n16, 4c + 2*half + {0,1}]
    v2f aV[4];
#pragma unroll
    for (int c = 0; c < 4; ++c) {
        const float2 t = *(const float2*)(V + n16 * 16 + 4 * c + 2 * half);
        aV[c].x = t.x;
        aV[c].y = t.y;
    }

#pragma unroll 1
    for (int t = 0; t < TILES_PER_WAVE; ++t) {
        const int base = (wave * TILES_PER_WAVE + t) * 16;
        if (base >= nbatch) break;

        const float4 nz = *(const float4*)(noise + (base + n16) * 4);
        float c0, s0, c1, s1, c2, s2, c3, s3;
        __sincosf(0.5f * nz.x, &s0, &c0);  // wire 0 -> bit 3
        __sincosf(0.5f * nz.y, &s1, &c1);  // wire 1 -> bit 2
        __sincosf(0.5f * nz.z, &s2, &c2);  // wire 2 -> bit 1
        __sincosf(0.5f * nz.w, &s3, &c3);  // wire 3 -> bit 0

        const float F0 = c0 * c1, F1 = c0 * s1, F2 = s0 * c1, F3 = s0 * s1;
        const float g2 = hb ? s2 : c2;   // bit1 of j equals `half`
        const float u0 = g2 * c3;        // d = 0
        const float u1 = g2 * s3;        // d = 1

        v8f dre  = {};
        v8f dim2 = {};
#pragma unroll
        for (int c = 0; c < 4; ++c) {
            const float F  = (c == 0) ? F0 : (c == 1) ? F1 : (c == 2) ? F2 : F3;
            const float r0 = F * u0;
            const float r1 = F * u1;
            const int   q0 = (c == 0) ? 0 : (c == 3) ? 2 : 1;  // popcount(c)
            const int   q1 = q0 + 1;

            const float sre0 = (q0 == 0 || q0 == 3) ? r0 : -r0;
            const float sim0 = (q0 <= 1) ? -r0 : r0;
            const float sre1 = (q1 == 0 || q1 == 3) ? r1 : -r1;
            const float sim1 = (q1 <= 1) ? -r1 : r1;

            v2f bre, bim;
            bre.x = (hb == ((q0 & 1) != 0)) ? sre0 : 0.0f;
            bim.x = (hb == ((q0 & 1) != 0)) ? 0.0f : sim0;
            bre.y = (hb == ((q1 & 1) != 0)) ? sre1 : 0.0f;
            bim.y = (hb == ((q1 & 1) != 0)) ? 0.0f : sim1;

            dre = __builtin_amdgcn_wmma_f32_16x16x4_f32(
                false, aV[c], false, bre, (short)0, dre, false, false);
            dim2 = __builtin_amdgcn_wmma_f32_16x16x4_f32(
                false, aV[c], false, bim, (short)0, dim2, false, false);
        }

        // probs: lane holds probs[M = g + 8*half, N = n16], g = 0..7
        float d[8];
#pragma unroll
        for (int g = 0; g < 8; ++g)
            d[g] = fmaf(dre[g], dre[g], dim2[g] * dim2[g]);

        float tw1  = (d[0] + d[1]) + (d[2] + d[3]);             // bit2(g)==0
        const float hi = (d[4] + d[5]) + (d[6] + d[7]);
        const float sAll = tw1 + hi;                            // wire0 (lanes<16)
        float tw2  = (d[0] + d[1]) + (d[4] + d[5]);             // bit1(g)==0
        float tw3  = (d[0] + d[2]) + (d[4] + d[6]);             // bit0(g)==0
        tw1 += swap_half_f32(tw1);
        tw2 += swap_half_f32(tw2);
        tw3 += swap_half_f32(tw3);

        if (half == 0) {
            float4 o;
            o.x = sAll;
            o.y = tw1;
            o.z = tw2;
            o.w = tw3;
            *(float4*)(out + (base + n16) * 4) = o;
        }
    }
}

extern "C" void kernel_launch(void* const* d_in, const int* in_sizes, int n_in,
                              void* d_out, int out_size, void* d_ws,
                              size_t ws_size, hipStream_t stream) {
    const float* noise   = (const float*)d_in[0];
    const float* weights = (const float*)d_in[1];
    float* out = (float*)d_out;
    float* V   = (float*)d_ws;  // 256 floats of scratch

    build_V_kernel<<<1, 16, 0, stream>>>(weights, V);

    const int N      = in_sizes[0] / N_QUBITS;  // batch size
    const int tiles  = (N + 15) / 16;
    const int waves  = (tiles + TILES_PER_WAVE - 1) / TILES_PER_WAVE;
    const int blocks = (waves + WAVES_PER_BLOCK - 1) / WAVES_PER_BLOCK;
    qcirc_wmma_kernel<<<blocks, WAVES_PER_BLOCK * 32, 0, stream>>>(noise, V, out, N);
}